// RWKV_RNN_23433341567179
// MI455X (gfx1250) — compile-verified
//
#include <hip/hip_runtime.h>
#include <hip/hip_bf16.h>

// ---------------------------------------------------------------------------
// RWKV-v4 single-token forward as ONE persistent megakernel on gfx1250.
//  - Bandwidth bound: 1.44 GB of fp32 weights / token, roofline ~62us @23.3TB/s
//  - Global phase barrier (agent-scope atomics) instead of ~170 kernel launches
//  - All matvecs through v_wmma_f32_16x16x4_f32 (exact fp32), weights streamed
//    with one fully-consumed nontemporal b128 load per lane per 2 WMMAs.
// ---------------------------------------------------------------------------

#define NL   24
#define D    1024
#define HD   4096
#define VSZ  32000
#define TPB  256
#define NWG  24

typedef float v2f __attribute__((ext_vector_type(2)));
typedef float v4f __attribute__((ext_vector_type(4)));
typedef float v8f __attribute__((ext_vector_type(8)));

struct Params {
  const int*   token;
  const float* st_in;
  const float* emb;
  const float* ln0_w; const float* ln0_b;
  const float* ln1_w; const float* ln1_b;
  const float* ln2_w; const float* ln2_b;
  const float* att_tmk; const float* att_tmv; const float* att_tmr;
  const float* time_first; const float* time_decay;
  const float* att_kw; const float* att_vw; const float* att_rw; const float* att_ow;
  const float* ffn_tmk; const float* ffn_tmr;
  const float* ffn_kw; const float* ffn_vw; const float* ffn_rw;
  const float* lnout_w; const float* lnout_b;
  const float* head;
  float* logits;   // d_out[0..V)
  float* st;       // d_out[V..V+5*NL*D)  (output state, updated in place)
  float* buf;      // scratch vectors in d_ws
  unsigned* cnt;   // global barrier counter in d_ws
};

// ---- device-wide phase barrier (all NWG workgroups resident) --------------
__device__ __forceinline__ void gbar(unsigned* cnt, unsigned& phase) {
  __threadfence();                 // release my stores to device scope
  __syncthreads();
  if (threadIdx.x == 0) {
    const unsigned target = (++phase) * gridDim.x;
    __hip_atomic_fetch_add(cnt, 1u, __ATOMIC_ACQ_REL, __HIP_MEMORY_SCOPE_AGENT);
    while (__hip_atomic_load(cnt, __ATOMIC_ACQUIRE, __HIP_MEMORY_SCOPE_AGENT) < target)
      __builtin_amdgcn_s_sleep(8);
  }
  __syncthreads();
  __threadfence();                 // acquire: discard stale cached lines
}

// ---- block-wide sum of two values, result broadcast -----------------------
__device__ __forceinline__ float2 block_reduce2(float a, float b) {
  __shared__ float sa[TPB / 32], sb[TPB / 32];
  const int lane = threadIdx.x & 31, w = threadIdx.x >> 5;
#pragma unroll
  for (int o = 16; o > 0; o >>= 1) { a += __shfl_xor(a, o, 32); b += __shfl_xor(b, o, 32); }
  __syncthreads();                 // protect sa/sb reuse across calls
  if (lane == 0) { sa[w] = a; sb[w] = b; }
  __syncthreads();
  float ra = 0.f, rb = 0.f;
#pragma unroll
  for (int i = 0; i < TPB / 32; ++i) { ra += sa[i]; rb += sb[i]; }
  return make_float2(ra, rb);
}

// ---- LayerNorm (+optional state mix) over D elems; WGs 0..3 participate ---
__device__ __forceinline__ void ln_mix(
    const float* __restrict__ xin, const float* __restrict__ w, const float* __restrict__ b,
    float* __restrict__ strow,
    const float* __restrict__ tm0, float* __restrict__ o0,
    const float* __restrict__ tm1, float* __restrict__ o1,
    const float* __restrict__ tm2, float* __restrict__ o2, int wg)
{
  if (wg * TPB >= D) return;       // uniform per workgroup
  float s = 0.f, s2 = 0.f;
  for (int i = threadIdx.x; i < D; i += TPB) { const float t = xin[i]; s += t; s2 += t * t; }
  const float2 r = block_reduce2(s, s2);
  const float mean = r.x * (1.f / D);
  const float inv  = rsqrtf(r.y * (1.f / D) - mean * mean + 1e-5f);
  const int i = wg * TPB + threadIdx.x;
  const float xn = (xin[i] - mean) * inv * w[i] + b[i];
  float prev = 0.f;
  if (strow) { prev = strow[i]; strow[i] = xn; }
  if (o0) o0[i] = tm0 ? (xn * tm0[i] + prev * (1.f - tm0[i])) : xn;
  if (o1) o1[i] = xn * tm1[i] + prev * (1.f - tm1[i]);
  if (o2) o2[i] = xn * tm2[i] + prev * (1.f - tm2[i]);
}

// ---- WMMA matvec: y[r0..r0+16) per wave, 128 rows per WG, strided over M --
// A = x-chunk broadcast into all 16 rows (from LDS), B = 16 weight rows.
// K permuted in groups of 8 so each lane's b128 weight load is fully used.
// mode: 0 plain, 1 y=add+dot, 2 y=relu(dot)^2, 3 y=add+sigmoid(gate)*dot
__device__ __forceinline__ void matvec_part(
    const float* __restrict__ W, const float* __restrict__ xsrc,
    float* __restrict__ y, int M, int K, int mode,
    const float* __restrict__ add, const float* __restrict__ gate,
    int wgl, int nwgp, float* xs)
{
  for (int i = threadIdx.x; i < K; i += TPB) xs[i] = xsrc[i];   // stage x in LDS
  __syncthreads();
  const int wave = threadIdx.x >> 5;
  const int lane = threadIdx.x & 31;
  const int half = lane >> 4;        // 0: k,k+1   1: k+4,k+5 (perm-matched)
  const int row  = lane & 15;
  for (int rb = wgl * 128; rb < M; rb += nwgp * 128) {
    const int r0 = rb + wave * 16;
    const float* wp = W + (size_t)(r0 + row) * K + half * 4;
    v8f c = {0.f, 0.f, 0.f, 0.f, 0.f, 0.f, 0.f, 0.f};
#pragma unroll 4
    for (int k0 = 0; k0 < K; k0 += 8) {
      const v4f wv = __builtin_nontemporal_load((const v4f*)(wp + k0));
      const float* xp = xs + k0 + half * 4;
      v2f b0 = { wv.x, wv.y };
      v2f b1 = { wv.z, wv.w };
      v2f a0 = { xp[0], xp[1] };
      v2f a1 = { xp[2], xp[3] };
      c = __builtin_amdgcn_wmma_f32_16x16x4_f32(false, a0, false, b0, (short)0, c, false, false);
      c = __builtin_amdgcn_wmma_f32_16x16x4_f32(false, a1, false, b1, (short)0, c, false, false);
    }
    if (lane < 16) {                 // D row M=0, lanes 0..15 = the 16 dots
      const int m = r0 + lane;
      float v = c[0];
      if      (mode == 1) v += add[m];
      else if (mode == 2) { v = fmaxf(v, 0.f); v *= v; }
      else if (mode == 3) v = add[m] + v / (1.f + __expf(-gate[m]));
      y[m] = v;
    }
  }
  __syncthreads();                   // xs reused by next stage
}

__global__ void init_cnt(unsigned* c) { *c = 0u; }

__global__ void __launch_bounds__(TPB, 1) rwkv_mega(Params p) {
  __shared__ float xs[HD];           // 16 KB x-staging (max K = 4096)
  const int wg = blockIdx.x;
  unsigned phase = 0;

  float* x   = p.buf;
  float* xk  = p.buf + 1 * D;
  float* xv  = p.buf + 2 * D;
  float* xr  = p.buf + 3 * D;
  float* kb  = p.buf + 4 * D;
  float* vb  = p.buf + 5 * D;
  float* rp  = p.buf + 6 * D;
  float* rwk = p.buf + 7 * D;
  float* rrp = p.buf + 8 * D;
  float* xf  = p.buf + 9 * D;
  float* kk  = p.buf + 10 * D;       // HD elems
  float* st  = p.st;

  // copy input state -> output state (operated on in place afterwards)
  for (int i = wg * TPB + threadIdx.x; i < 5 * NL * D; i += NWG * TPB) st[i] = p.st_in[i];
  // x = LN(emb[token], ln0)
  const int tok = p.token[0];
  ln_mix(p.emb + (size_t)tok * D, p.ln0_w, p.ln0_b, nullptr,
         nullptr, x, nullptr, nullptr, nullptr, nullptr, wg);
  gbar(p.cnt, phase);

  for (int l = 0; l < NL; ++l) {
    const size_t o1 = (size_t)l * D;
    const size_t o2 = (size_t)l * D * D;
    const size_t o3 = (size_t)l * HD * D;

    // ---- time mixing ----
    ln_mix(x, p.ln1_w + o1, p.ln1_b + o1, st + (size_t)(5 * l + 1) * D,
           p.att_tmk + o1, xk, p.att_tmv + o1, xv, p.att_tmr + o1, xr, wg);
    gbar(p.cnt, phase);

    if      (wg < 8)  matvec_part(p.att_rw + o2, xr, rp, D, D, 0, nullptr, nullptr, wg,      8, xs);
    else if (wg < 16) matvec_part(p.att_kw + o2, xk, kb, D, D, 0, nullptr, nullptr, wg - 8,  8, xs);
    else              matvec_part(p.att_vw + o2, xv, vb, D, D, 0, nullptr, nullptr, wg - 16, 8, xs);
    gbar(p.cnt, phase);

    { // wkv + state update (elementwise, WGs 0..3)
      const int i = wg * TPB + threadIdx.x;
      if (i < D) {
        const float kv = kb[i], vv = vb[i];
        const float aa = st[(size_t)(5 * l + 2) * D + i];
        const float bb = st[(size_t)(5 * l + 3) * D + i];
        const float pp = st[(size_t)(5 * l + 4) * D + i];
        const float ww = p.time_first[o1 + i] + kv;
        const float qq = fmaxf(pp, ww);
        const float e1 = __expf(pp - qq), e2 = __expf(ww - qq);
        const float wkv = (e1 * aa + e2 * vv) / (e1 * bb + e2);
        rwk[i] = wkv / (1.f + __expf(-rp[i]));          // sigmoid(r) * wkv
        const float ww2 = pp + p.time_decay[o1 + i];
        const float qq2 = fmaxf(ww2, kv);
        const float e1b = __expf(ww2 - qq2), e2b = __expf(kv - qq2);
        st[(size_t)(5 * l + 2) * D + i] = e1b * aa + e2b * vv;
        st[(size_t)(5 * l + 3) * D + i] = e1b * bb + e2b;
        st[(size_t)(5 * l + 4) * D + i] = qq2;
      }
    }
    gbar(p.cnt, phase);

    if (wg < 8) matvec_part(p.att_ow + o2, rwk, x, D, D, 1, x, nullptr, wg, 8, xs); // x += ow@rwkv
    gbar(p.cnt, phase);

    // ---- channel mixing ----
    ln_mix(x, p.ln2_w + o1, p.ln2_b + o1, st + (size_t)(5 * l) * D,
           p.ffn_tmk + o1, xk, p.ffn_tmr + o1, xr, nullptr, nullptr, wg);
    gbar(p.cnt, phase);

    if (wg < 16) matvec_part(p.ffn_kw + o3, xk, kk,  HD, D, 2, nullptr, nullptr, wg,      16, xs);
    else         matvec_part(p.ffn_rw + o2, xr, rrp, D,  D, 0, nullptr, nullptr, wg - 16, 8,  xs);
    gbar(p.cnt, phase);

    if (wg < 8) matvec_part(p.ffn_vw + o3, kk, x, D, HD, 3, x, rrp, wg, 8, xs); // x += sig(rr)*vw@kk
    gbar(p.cnt, phase);
  }

  ln_mix(x, p.lnout_w, p.lnout_b, nullptr, nullptr, xf,
         nullptr, nullptr, nullptr, nullptr, wg);
  gbar(p.cnt, phase);

  matvec_part(p.head, xf, p.logits, VSZ, D, 0, nullptr, nullptr, wg, NWG, xs);
}

extern "C" void kernel_launch(void* const* d_in, const int* in_sizes, int n_in,
                              void* d_out, int out_size, void* d_ws, size_t ws_size,
                              hipStream_t stream) {
  (void)in_sizes; (void)n_in; (void)out_size; (void)ws_size;
  Params p;
  p.token      = (const int*)  d_in[0];
  p.st_in      = (const float*)d_in[1];
  p.emb        = (const float*)d_in[2];
  p.ln0_w      = (const float*)d_in[3];  p.ln0_b = (const float*)d_in[4];
  p.ln1_w      = (const float*)d_in[5];  p.ln1_b = (const float*)d_in[6];
  p.ln2_w      = (const float*)d_in[7];  p.ln2_b = (const float*)d_in[8];
  p.att_tmk    = (const float*)d_in[9];
  p.att_tmv    = (const float*)d_in[10];
  p.att_tmr    = (const float*)d_in[11];
  p.time_first = (const float*)d_in[12];
  p.time_decay = (const float*)d_in[13];
  p.att_kw     = (const float*)d_in[14];
  p.att_vw     = (const float*)d_in[15];
  p.att_rw     = (const float*)d_in[16];
  p.att_ow     = (const float*)d_in[17];
  p.ffn_tmk    = (const float*)d_in[18];
  p.ffn_tmr    = (const float*)d_in[19];
  p.ffn_kw     = (const float*)d_in[20];
  p.ffn_vw     = (const float*)d_in[21];
  p.ffn_rw     = (const float*)d_in[22];
  p.lnout_w    = (const float*)d_in[23]; p.lnout_b = (const float*)d_in[24];
  p.head       = (const float*)d_in[25];
  p.logits     = (float*)d_out;
  p.st         = (float*)d_out + VSZ;
  p.cnt        = (unsigned*)d_ws;
  p.buf        = (float*)d_ws + 256;     // ~(10*D + HD)*4B ≈ 57 KB scratch

  init_cnt<<<1, 1, 0, stream>>>(p.cnt);
  rwkv_mega<<<dim3(NWG), dim3(TPB), 0, stream>>>(p);
}